// TrajectoryFreeLoss_19181323944326
// MI455X (gfx1250) — compile-verified
//
#include <hip/hip_runtime.h>
#include <math.h>

#define NPART   512
#define LSNAP   8
#define NSNAP   32      // M*L
#define HDIM    32
#define NSTEP   28      // M*(L-1)
#define SIGMA2  0.01f
#define EPSCL   1e-20f
#define HO      (NSNAP * NPART)   // offset between the two i-half partials

typedef float v2f __attribute__((ext_vector_type(2)));
typedef float v8f __attribute__((ext_vector_type(8)));

__device__ __forceinline__ float tanh_fast(float x) {
#if __has_builtin(__builtin_amdgcn_tanhf)
    return __builtin_amdgcn_tanhf(x);   // gfx1250 v_tanh_f32
#else
    return tanhf(x);
#endif
}

__device__ __forceinline__ float rsq_fast(float x) {
#if __has_builtin(__builtin_amdgcn_rsqf)
    return __builtin_amdgcn_rsqf(x);    // v_rsq_f32
#else
    return 1.0f / sqrtf(x);
#endif
}

// ---------------- K1: per-particle V terms (all 32 snapshots) ----------------
__global__ void vterms_kernel(const float* __restrict__ data,
                              const float* __restrict__ W1,
                              const float* __restrict__ b1,
                              const float* __restrict__ w2,
                              float* __restrict__ gV,
                              float* __restrict__ lapV,
                              float* __restrict__ VPart) {
    __shared__ float red[256];
    const int tid = threadIdx.x;
    const int gid = blockIdx.x * 256 + tid;           // gid = s*512 + i
    const float x = data[gid * 2 + 0];
    const float y = data[gid * 2 + 1];
    float V = 0.f, gx = 0.f, gy = 0.f, lap = 0.f;
    #pragma unroll 8
    for (int h = 0; h < HDIM; ++h) {
        const float w1x = W1[h], w1y = W1[HDIM + h];
        const float z  = fmaf(x, w1x, fmaf(y, w1y, b1[h]));
        const float t  = tanh_fast(z);
        const float s1 = fmaf(-t, t, 1.0f);           // sech^2
        const float g  = s1 * w2[h];
        V   = fmaf(t, w2[h], V);
        gx  = fmaf(g, w1x, gx);
        gy  = fmaf(g, w1y, gy);
        lap = fmaf(t * g, -2.0f * fmaf(w1x, w1x, w1y * w1y), lap);
    }
    gV[gid * 2 + 0] = gx;
    gV[gid * 2 + 1] = gy;
    lapV[gid] = lap;
    red[tid] = V;
    __syncthreads();
    for (int off = 128; off > 0; off >>= 1) {
        if (tid < off) red[tid] += red[tid + off];
        __syncthreads();
    }
    if (tid == 0) VPart[blockIdx.x] = red[0];         // 2 partials per snapshot
}

// ---------------- K2: pairwise kernel (WMMA distance tiles + tanh) ----------
// Wave owns (snapshot s, 16-column j-tile, i-half). 16 i-tiles per wave.
// sq_ij via augmented K=4 WMMA: (xi, yi, |xi|^2, 1) . (-2xj, -2yj, 1, |xj|^2).
__global__ void __launch_bounds__(256)
pair_kernel(const float* __restrict__ data,
            const float* __restrict__ p1,
            const float* __restrict__ pb1,
            const float* __restrict__ p2,
            float* __restrict__ Aout,   // [2][NSNAP][NPART] partials
            float* __restrict__ Cout,
            float* __restrict__ Dxout,
            float* __restrict__ Dyout,
            float* __restrict__ phiPart) {  // [2][NSNAP][32]
    __shared__ float4 s4[HDIM];                       // (p1, pb1, p1*p2, p2)
    const int tid = threadIdx.x;
    if (tid < HDIM) {
        const float a = p1[tid], b = pb1[tid], c = p2[tid];
        s4[tid] = make_float4(a, b, a * c, c);
    }
    __syncthreads();

    const int lane  = tid & 31;
    const int wave  = tid >> 5;
    const int s     = blockIdx.x >> 3;                // snapshot 0..31
    const int sub   = blockIdx.x & 7;
    const int ihalf = sub & 1;                        // which 16 i-tiles
    const int jtile = ((sub >> 1) << 3) + wave;       // 0..31
    const int j0    = jtile << 4;
    const int ml    = lane & 15;
    const int half  = (lane >> 4) << 3;               // rows 0..7 or 8..15
    const float* X  = data + s * (NPART * 2);

    const int   jn = j0 + ml;
    const float xj = X[jn * 2 + 0];
    const float yj = X[jn * 2 + 1];
    v2f bfrag;
    if (lane < 16) { bfrag[0] = -2.0f * xj; bfrag[1] = -2.0f * yj; }          // K0,K1
    else           { bfrag[0] = 1.0f;       bfrag[1] = fmaf(xj, xj, yj * yj); } // K2,K3

    float AAcc = 0.f, CAcc = 0.f, DxAcc = 0.f, DyAcc = 0.f, phiAcc = 0.f;

    const int itEnd = (ihalf << 4) + 16;
    for (int it = (ihalf << 4); it < itEnd; ++it) {
        const int   i0 = it << 4;
        const float xi = X[(i0 + ml) * 2 + 0];
        const float yi = X[(i0 + ml) * 2 + 1];
        v2f afrag;
        if (lane < 16) { afrag[0] = xi; afrag[1] = yi; }                      // K0,K1
        else           { afrag[0] = fmaf(xi, xi, yi * yi); afrag[1] = 1.0f; } // K2,K3
        v8f cz = {0.f, 0.f, 0.f, 0.f, 0.f, 0.f, 0.f, 0.f};
        v8f sq = __builtin_amdgcn_wmma_f32_16x16x4_f32(
            false, afrag, false, bfrag, (short)0, cz, false, false);

        #pragma unroll
        for (int v = 0; v < 8; ++v) {
            const int   Mrow = v + half;
            const int   gi   = i0 + Mrow;
            const float sqc  = fmaxf(sq[v], EPSCL);
            const float rinv = rsq_fast(sqc);         // 1/r
            const float dist = sqc * rinv;            // r
            float phi = 0.f, dphi = 0.f, u = 0.f;     // u = Sum t*s*p1^2*p2
            #pragma unroll 8
            for (int h = 0; h < HDIM; ++h) {
                const float4 q = s4[h];
                const float arg = fmaf(dist, q.x, q.y);
                const float t   = tanh_fast(arg);
                const float s1  = fmaf(-t, t, 1.0f);
                phi  = fmaf(t,  q.w, phi);
                dphi = fmaf(s1, q.z, dphi);
                u    = fmaf(t * s1 * q.x, q.z, u);
            }
            const float msk = (gi == jn) ? 0.0f : 1.0f;
            const float w   = dphi * rinv;
            const float xs  = __shfl(xi, Mrow, 32);   // x_i of row Mrow
            const float ys  = __shfl(yi, Mrow, 32);
            AAcc   = fmaf(msk, w, AAcc);
            CAcc   = fmaf(msk, fmaf(-2.0f, u, w), CAcc);
            phiAcc = fmaf(msk, phi, phiAcc);
            const float mw = msk * w;
            DxAcc  = fmaf(mw, xs, DxAcc);
            DyAcc  = fmaf(mw, ys, DyAcc);
        }
    }
    // combine the two half-rows of each column (lanes L and L+16)
    AAcc  += __shfl_xor(AAcc,  16, 32);
    CAcc  += __shfl_xor(CAcc,  16, 32);
    DxAcc += __shfl_xor(DxAcc, 16, 32);
    DyAcc += __shfl_xor(DyAcc, 16, 32);
    if (lane < 16) {
        const int o = ihalf * HO + s * NPART + j0 + lane;
        Aout[o] = AAcc; Cout[o] = CAcc; Dxout[o] = DxAcc; Dyout[o] = DyAcc;
    }
    for (int off = 16; off > 0; off >>= 1) phiAcc += __shfl_xor(phiAcc, off, 32);
    if (lane == 0) phiPart[ihalf * (NSNAP * 32) + s * 32 + jtile] = phiAcc;
}

// ---------------- K3: finalize to scalar ------------------------------------
__global__ void finalize_kernel(const float* __restrict__ data,
                                const float* __restrict__ tsnap,
                                const float* __restrict__ b2,
                                const float* __restrict__ pb2,
                                const float* __restrict__ gV,
                                const float* __restrict__ lapV,
                                const float* __restrict__ Aacc,
                                const float* __restrict__ Cacc,
                                const float* __restrict__ Dx,
                                const float* __restrict__ Dy,
                                const float* __restrict__ VPart,
                                const float* __restrict__ phiPart,
                                float* __restrict__ out) {
    __shared__ float r1[256], r2[256];
    const int tid = threadIdx.x;
    const float invN = 1.0f / (float)NPART;
    float sdiss = 0.f, sdiff = 0.f;
    for (int k = tid; k < NSTEP * NPART; k += 256) {
        const int step = k >> 9;
        const int i    = k & 511;
        const int m    = step / (LSNAP - 1);
        const int l    = step % (LSNAP - 1);
        const int sc   = m * LSNAP + l;
        const float dt = tsnap[l + 1] - tsnap[l];
        const int base = sc * NPART + i;
        const float xi = data[base * 2 + 0], yi = data[base * 2 + 1];
        const float Ai = Aacc[base] + Aacc[HO + base];
        const float Dxi = Dx[base] + Dx[HO + base];
        const float Dyi = Dy[base] + Dy[HO + base];
        const float gpx = (xi * Ai - Dxi) * invN;
        const float gpy = (yi * Ai - Dyi) * invN;
        const float dx = -gV[base * 2 + 0] - gpx;
        const float dy = -gV[base * 2 + 1] - gpy;
        sdiss = fmaf(fmaf(dx, dx, dy * dy), dt, sdiss);
        const float Ci = Cacc[base] + Cacc[HO + base];
        const float lap = lapV[base] + Ci * invN;
        sdiff = fmaf(lap, dt, sdiff);
    }
    r1[tid] = sdiss; r2[tid] = sdiff;
    __syncthreads();
    for (int off = 128; off > 0; off >>= 1) {
        if (tid < off) { r1[tid] += r1[tid + off]; r2[tid] += r2[tid + off]; }
        __syncthreads();
    }
    if (tid == 0) {
        const float Jdiss = r1[0] * invN;
        const float Jdiff = SIGMA2 * r2[0] * invN;
        const float invN2 = invN * invN;
        const float cphi  = pb2[0] * (float)(NPART * NPART - NPART);
        float E[NSNAP];
        for (int si = 0; si < NSNAP; ++si) {
            float ps = 0.f;
            for (int jt = 0; jt < 32; ++jt)
                ps += phiPart[si * 32 + jt] + phiPart[NSNAP * 32 + si * 32 + jt];
            E[si] = (VPart[si * 2] + VPart[si * 2 + 1]) * invN + b2[0]
                  + (ps + cphi) * invN2;
        }
        float dE = 0.f;
        for (int step = 0; step < NSTEP; ++step) {
            const int m = step / (LSNAP - 1), l = step % (LSNAP - 1);
            const int sc = m * LSNAP + l;
            dE += E[sc + 1] - E[sc];
        }
        const float invS = 1.0f / (float)NSTEP;
        const float t0 = Jdiss * invS, t1 = Jdiff * invS, t2 = dE * invS;
        const float r = t0 + t1 - 2.0f * t2;
        out[0] = r * r;
    }
}

extern "C" void kernel_launch(void* const* d_in, const int* in_sizes, int n_in,
                              void* d_out, int out_size, void* d_ws, size_t ws_size,
                              hipStream_t stream) {
    const float* data  = (const float*)d_in[0];
    const float* tsnap = (const float*)d_in[1];
    const float* W1    = (const float*)d_in[2];
    const float* b1    = (const float*)d_in[3];
    const float* w2    = (const float*)d_in[4];
    const float* b2    = (const float*)d_in[5];
    const float* p1    = (const float*)d_in[6];
    const float* pb1   = (const float*)d_in[7];
    const float* p2    = (const float*)d_in[8];
    const float* pb2   = (const float*)d_in[9];

    float* ws      = (float*)d_ws;
    float* gV      = ws;                          // 32*512*2
    float* lapV    = gV   + NSNAP * NPART * 2;    // 32*512
    float* Aacc    = lapV + NSNAP * NPART;        // 2*32*512 each below
    float* Cacc    = Aacc + 2 * NSNAP * NPART;
    float* Dx      = Cacc + 2 * NSNAP * NPART;
    float* Dy      = Dx   + 2 * NSNAP * NPART;
    float* VPart   = Dy   + 2 * NSNAP * NPART;    // 64
    float* phiPart = VPart + 64;                  // 2*32*32

    vterms_kernel<<<64, 256, 0, stream>>>(data, W1, b1, w2, gV, lapV, VPart);
    pair_kernel<<<NSNAP * 8, 256, 0, stream>>>(data, p1, pb1, p2,
                                               Aacc, Cacc, Dx, Dy, phiPart);
    finalize_kernel<<<1, 256, 0, stream>>>(data, tsnap, b2, pb2, gV, lapV,
                                           Aacc, Cacc, Dx, Dy, VPart, phiPart,
                                           (float*)d_out);
}